// Environment_22634477650530
// MI455X (gfx1250) — compile-verified
//
#include <hip/hip_runtime.h>

typedef float v2f __attribute__((ext_vector_type(2)));
typedef float v8f __attribute__((ext_vector_type(8)));

#define THREADS 256
#define ROWS_PER_THREAD 16
#define ROWS_PER_BLOCK (THREADS * ROWS_PER_THREAD) /* 4096 */
#define ROWS_PER_WAVE (32 * ROWS_PER_THREAD)       /* 512 */

// ---------------------------------------------------------------------------
// Exclusive prefix scan of two independent 8-float sequences (xa, xb) using
// V_WMMA_F32_16X16X4_F32.  y = L_blockdiag_strict * x computed as 4 chained
// WMMAs (K-chunks of 4).  Data + strict-lower block-diagonal mask live in the
// A operand (16x4 f32 layout: lane m holds A[m][0..1] in VGPR0/1, lane 16+m
// holds A[m][2..3]).  B is the all-ones 4x16 matrix, which is invariant under
// any lane/VGPR layout permutation, so no B-layout assumption is needed.
// D layout: lane 0 holds y_0..y_7 in acc[0..7], lane 16
// holds y_8..y_15.  Must be executed by one full wave (EXEC all ones).
// ---------------------------------------------------------------------------
static __device__ __forceinline__ void dual_scan8_excl(const float* xa, const float* xb,
                                                       float* oa, float* ob) {
    const int lane = (int)(threadIdx.x & 31u);
    const int m    = lane & 15;
    const int half = lane >> 4;
    v8f acc = {0.f, 0.f, 0.f, 0.f, 0.f, 0.f, 0.f, 0.f};
    const v2f bones = {1.0f, 1.0f};
#pragma unroll
    for (int cc = 0; cc < 4; ++cc) {
        const int k0 = cc * 4 + half * 2;
        const int k1 = k0 + 1;
        const float x0 = (k0 < 8) ? xa[k0] : xb[k0 - 8];
        const float x1 = (k1 < 8) ? xa[k1] : xb[k1 - 8];
        v2f a;
        a.x = ((k0 < m) && ((k0 >> 3) == (m >> 3))) ? x0 : 0.f;
        a.y = ((k1 < m) && ((k1 >> 3) == (m >> 3))) ? x1 : 0.f;
        acc = __builtin_amdgcn_wmma_f32_16x16x4_f32(false, a, false, bones,
                                                    (short)0, acc, false, false);
    }
    if (lane == 0) {
#pragma unroll
        for (int r = 0; r < 8; ++r) oa[r] = acc[r];
    } else if (lane == 16) {
#pragma unroll
        for (int r = 0; r < 8; ++r) ob[r] = acc[r];
    }
}

// ---------------------------------------------------------------------------
// Pass 1: per-block aggregates of the force increments g_i = (DT/M)*f[row].
//   T1 = sum g_i              (i = block-local step index, 0..C-1)
//   T2 = sum (C - i) * g_i    ( = sum over block of block-local first prefix )
// Step i (global n) reads f[max(n-1,0)]  (reference index quirk).
// ---------------------------------------------------------------------------
__global__ void phase1_aggregate(const float* __restrict__ f, float* __restrict__ agg,
                                 long K) {
    const long s = (long)blockIdx.x * ROWS_PER_BLOCK;
    const int tid = (int)threadIdx.x;
    const float SCALE = (float)(0.01 / 1.5);
    float T1[3] = {0.f, 0.f, 0.f}, T2[3] = {0.f, 0.f, 0.f};
#pragma unroll
    for (int j = 0; j < ROWS_PER_THREAD; ++j) {
        const int i = tid + j * THREADS; /* local step index: coalesced */
        long r = s + i - 1;
        if (r < 0) r = 0;
        if (r > K - 1) r = K - 1;
        const float w = (float)(ROWS_PER_BLOCK - i);
        const float gx = SCALE * f[3 * r + 0];
        const float gy = SCALE * f[3 * r + 1];
        const float gz = SCALE * f[3 * r + 2];
        T1[0] += gx; T1[1] += gy; T1[2] += gz;
        T2[0] += w * gx; T2[1] += w * gy; T2[2] += w * gz;
    }
    __shared__ float red[THREADS][6];
    red[tid][0] = T1[0]; red[tid][1] = T1[1]; red[tid][2] = T1[2];
    red[tid][3] = T2[0]; red[tid][4] = T2[1]; red[tid][5] = T2[2];
    __syncthreads();
    for (int st = THREADS / 2; st > 0; st >>= 1) {
        if (tid < st) {
#pragma unroll
            for (int c = 0; c < 6; ++c) red[tid][c] += red[tid + st][c];
        }
        __syncthreads();
    }
    if (tid == 0) {
#pragma unroll
        for (int c = 0; c < 6; ++c) agg[(size_t)blockIdx.x * 6 + c] = red[0][c];
    }
}

// ---------------------------------------------------------------------------
// Pass 2: single block.  f64 tuple scan over block aggregates.
// Tuple (t1,t2,rows): combining X (earlier) then Y: t2 = t2x + t2y + rows_y*t1x.
// Emits per-block exclusive carries: carry1 = G1 before block, carry2 = G2
// before block (force-only), in double.
// ---------------------------------------------------------------------------
__global__ void phase2_scan(const float* __restrict__ agg, double* __restrict__ carry,
                            int NB) {
    const int tid = (int)threadIdx.x;
    const int lane = tid & 31;
    const int wave = tid >> 5;
    const int PER = (NB + THREADS - 1) / THREADS;
    const double Crows = (double)ROWS_PER_BLOCK;

    // Serial local tuple over this thread's PER consecutive blocks.
    double t1[3] = {0, 0, 0}, t2[3] = {0, 0, 0};
    for (int j = 0; j < PER; ++j) {
        const int b = tid * PER + j;
        double a1[3] = {0, 0, 0}, a2[3] = {0, 0, 0};
        if (b < NB) {
#pragma unroll
            for (int c = 0; c < 3; ++c) {
                a1[c] = (double)agg[(size_t)b * 6 + c];
                a2[c] = (double)agg[(size_t)b * 6 + 3 + c];
            }
        }
#pragma unroll
        for (int c = 0; c < 3; ++c) { t2[c] += a2[c] + Crows * t1[c]; }
#pragma unroll
        for (int c = 0; c < 3; ++c) { t1[c] += a1[c]; }
    }
    const double ownRows = (double)PER * Crows;
    double o1[3], o2[3];
#pragma unroll
    for (int c = 0; c < 3; ++c) { o1[c] = t1[c]; o2[c] = t2[c]; }

    // Inclusive wave32 tuple scan.
    double rowsAcc = ownRows;
    for (int d = 1; d < 32; d <<= 1) {
        double b1[3], b2[3], brows;
#pragma unroll
        for (int c = 0; c < 3; ++c) {
            b1[c] = __shfl_up(t1[c], (unsigned)d, 32);
            b2[c] = __shfl_up(t2[c], (unsigned)d, 32);
        }
        brows = __shfl_up(rowsAcc, (unsigned)d, 32);
        if (lane >= d) {
#pragma unroll
            for (int c = 0; c < 3; ++c) { t2[c] = b2[c] + t2[c] + rowsAcc * b1[c]; }
#pragma unroll
            for (int c = 0; c < 3; ++c) { t1[c] += b1[c]; }
            rowsAcc += brows;
        }
    }
    // Lane-exclusive tuple within wave.
    double l1[3], l2[3];
#pragma unroll
    for (int c = 0; c < 3; ++c) {
        l1[c] = t1[c] - o1[c];
        l2[c] = t2[c] - o2[c] - ownRows * l1[c];
    }
    __shared__ double wt1[8][3], wt2[8][3];
    if (lane == 31) {
#pragma unroll
        for (int c = 0; c < 3; ++c) { wt1[wave][c] = t1[c]; wt2[wave][c] = t2[c]; }
    }
    __syncthreads();
    // Wave-exclusive context (serial over <=7 entries).
    double wc1[3] = {0, 0, 0}, wc2[3] = {0, 0, 0};
    const double waveRows = 32.0 * ownRows;
    for (int w = 0; w < wave; ++w) {
#pragma unroll
        for (int c = 0; c < 3; ++c) { wc2[c] += wt2[w][c] + waveRows * wc1[c]; }
#pragma unroll
        for (int c = 0; c < 3; ++c) { wc1[c] += wt1[w][c]; }
    }
    // Full thread-exclusive tuple, then serially emit carries for own blocks.
    double r1[3], r2[3];
#pragma unroll
    for (int c = 0; c < 3; ++c) {
        r1[c] = wc1[c] + l1[c];
        r2[c] = wc2[c] + l2[c] + ((double)lane * ownRows) * wc1[c];
    }
    for (int j = 0; j < PER; ++j) {
        const int b = tid * PER + j;
        if (b < NB) {
#pragma unroll
            for (int c = 0; c < 3; ++c) {
                carry[(size_t)b * 6 + c] = r1[c];
                carry[(size_t)b * 6 + 3 + c] = r2[c];
            }
            double a1[3], a2[3];
#pragma unroll
            for (int c = 0; c < 3; ++c) {
                a1[c] = (double)agg[(size_t)b * 6 + c];
                a2[c] = (double)agg[(size_t)b * 6 + 3 + c];
            }
#pragma unroll
            for (int c = 0; c < 3; ++c) { r2[c] += a2[c] + Crows * r1[c]; }
#pragma unroll
            for (int c = 0; c < 3; ++c) { r1[c] += a1[c]; }
        }
    }
}

// ---------------------------------------------------------------------------
// Pass 3: re-scan each block with carries applied; emit p and v.
//   v_n = v0 + A_n ;  p_n = p0 + DT*(n+1)*v0 + DT*B_n - (DT/2)*A_n
// Gravity handled analytically:  A grav z = -(n+1)*DT*G ,
//                                B grav z = -DT*G*(n+1)(n+2)/2.
// ---------------------------------------------------------------------------
__global__ void phase3_emit(const float* __restrict__ f, const float* __restrict__ p0,
                            const float* __restrict__ v0,
                            const double* __restrict__ carry, float* __restrict__ out,
                            long K) {
    __shared__ float lds_f[3 * ROWS_PER_BLOCK]; /* 48 KB staging */
    __shared__ float WT1s[24], WT2s[24], E1[24], E2[24], EE[24];
    __shared__ float WC1[24], WC2[24], Zs[8], Junk8[8];
    const int tid = (int)threadIdx.x;
    const int lane = tid & 31;
    const int wave = tid >> 5;
    const long s = (long)blockIdx.x * ROWS_PER_BLOCK;
    const float SCALE = (float)(0.01 / 1.5);

    // Stage rows s-1 .. s+C-2 (step n reads f[n-1]; row -1 clamps to row 0).
    const long base = 3 * s - 3;
#pragma unroll
    for (int it = 0; it < (3 * ROWS_PER_BLOCK) / THREADS; ++it) {
        const int j = it * THREADS + tid;
        long g = base + (long)j;
        if (g < 0) g += 3; /* row -1, comp c -> row 0, comp c */
        if (g > 3 * K - 1) g = 3 * K - 1;
        lds_f[j] = f[g];
    }
    if (tid < 8) Zs[tid] = 0.f;
    __syncthreads();

    // Pass A: per-thread segment totals T1, T2 = sum of within-segment prefix.
    float T1[3] = {0.f, 0.f, 0.f}, T2[3] = {0.f, 0.f, 0.f};
#pragma unroll
    for (int q = 0; q < ROWS_PER_THREAD; ++q) {
        const int b3 = 3 * (tid * ROWS_PER_THREAD + q);
#pragma unroll
        for (int c = 0; c < 3; ++c) {
            T1[c] += SCALE * lds_f[b3 + c];
            T2[c] += T1[c];
        }
    }
    // Inclusive wave32 tuple scan (f32; len starts at R=16 rows).
    float a1[3] = {T1[0], T1[1], T1[2]}, a2[3] = {T2[0], T2[1], T2[2]};
    float rows = (float)ROWS_PER_THREAD;
    for (int d = 1; d < 32; d <<= 1) {
        float b1[3], b2[3], brows;
#pragma unroll
        for (int c = 0; c < 3; ++c) {
            b1[c] = __shfl_up(a1[c], (unsigned)d, 32);
            b2[c] = __shfl_up(a2[c], (unsigned)d, 32);
        }
        brows = __shfl_up(rows, (unsigned)d, 32);
        if (lane >= d) {
#pragma unroll
            for (int c = 0; c < 3; ++c) { a2[c] = b2[c] + a2[c] + rows * b1[c]; }
#pragma unroll
            for (int c = 0; c < 3; ++c) { a1[c] += b1[c]; }
            rows += brows;
        }
    }
    float l1[3], l2[3]; /* lane-exclusive tuple within wave */
#pragma unroll
    for (int c = 0; c < 3; ++c) {
        l1[c] = a1[c] - T1[c];
        l2[c] = a2[c] - T2[c] - (float)ROWS_PER_THREAD * l1[c];
    }
    if (lane == 31) {
#pragma unroll
        for (int c = 0; c < 3; ++c) { WT1s[c * 8 + wave] = a1[c]; WT2s[c * 8 + wave] = a2[c]; }
    }
    __syncthreads();

    // Level-2 combine across the 8 waves via WMMA exclusive 8-scans:
    //   e1 = excl(WT1), e2 = excl(WT2), ee1 = excl(e1);
    //   wave carry: wc1 = e1, wc2 = e2 + 512*ee1.
    if (wave == 0) {
        dual_scan8_excl(WT1s + 0, WT1s + 8, E1 + 0, E1 + 8);
        dual_scan8_excl(WT1s + 16, WT2s + 0, E1 + 16, E2 + 0);
        dual_scan8_excl(WT2s + 8, WT2s + 16, E2 + 8, E2 + 16);
        dual_scan8_excl(E1 + 0, E1 + 8, EE + 0, EE + 8);
        dual_scan8_excl(E1 + 16, Zs, EE + 16, Junk8);
        if (lane < 24) {
            const int w = lane & 7, c = lane >> 3;
            WC1[c * 8 + w] = E1[c * 8 + w];
            WC2[c * 8 + w] = E2[c * 8 + w] + (float)ROWS_PER_WAVE * EE[c * 8 + w];
        }
    }
    __syncthreads();

    // Block-exclusive context for this thread.
    float thE1[3], thW[3];
#pragma unroll
    for (int c = 0; c < 3; ++c) {
        const float w1 = WC1[c * 8 + wave], w2 = WC2[c * 8 + wave];
        thE1[c] = w1 + l1[c];
        thW[c] = w2 + l2[c] + ((float)ROWS_PER_THREAD * (float)lane) * w1;
    }

    // Per-thread f64 bases, then all per-element math in f32.
    const long N0 = s + (long)tid * ROWS_PER_THREAD;
    double c1d[3], c2d[3];
#pragma unroll
    for (int c = 0; c < 3; ++c) {
        c1d[c] = carry[(size_t)blockIdx.x * 6 + c];
        c2d[c] = carry[(size_t)blockIdx.x * 6 + 3 + c];
    }
    const double DTG = 0.01 * 9.81;
    float Af[3], Bf[3], Pf[3], dPf[3], c1f[3], v0f[3], p_dummy;
    (void)p_dummy;
#pragma unroll
    for (int c = 0; c < 3; ++c) {
        double A0 = c1d[c] + (double)thE1[c];
        double B0 = c2d[c] + (16.0 * (double)tid + 1.0) * c1d[c] + (double)thW[c] +
                    (double)thE1[c];
        if (c == 2) {
            A0 -= DTG * (double)(N0 + 1);
            B0 -= DTG * 0.5 * (double)(N0 + 1) * (double)(N0 + 2);
        }
        Af[c] = (float)A0;
        Bf[c] = (float)B0;
        const double v0d = (double)v0[c];
        Pf[c] = (float)((double)p0[c] + 0.01 * (double)(N0 + 1) * v0d);
        dPf[c] = (float)(0.01 * v0d);
        c1f[c] = (float)c1d[c];
        v0f[c] = v0[c];
    }
    const float dAz = -(float)DTG;
    const float gB1 = -(float)(DTG * 0.5 * (double)(2 * N0 + 3));
    const float gB2 = -(float)(DTG * 0.5);
    const float DTf = 0.01f;

    float run1[3] = {0.f, 0.f, 0.f}, run2[3] = {0.f, 0.f, 0.f};
    float* __restrict__ outp = out;
    float* __restrict__ outv = out + (size_t)3 * K;
#pragma unroll
    for (int q = 0; q < ROWS_PER_THREAD; ++q) {
        const long n = N0 + q;
        const int b3 = 3 * (tid * ROWS_PER_THREAD + q);
        const float qf = (float)q;
        if (n < K) {
#pragma unroll
            for (int c = 0; c < 3; ++c) {
                const float gv = SCALE * lds_f[b3 + c];
                run1[c] += gv;
                run2[c] += run1[c];
                float A = Af[c] + run1[c];
                float B2v = Bf[c] + qf * (c1f[c] + thE1[c]) + run2[c];
                if (c == 2) {
                    A += dAz * qf;
                    B2v += gB1 * qf + gB2 * qf * qf;
                }
                const float vv = v0f[c] + A;
                const float pp = Pf[c] + qf * dPf[c] + DTf * B2v - 0.5f * DTf * A;
                outp[3 * n + c] = pp;
                outv[3 * n + c] = vv;
            }
        }
    }
}

extern "C" void kernel_launch(void* const* d_in, const int* in_sizes, int n_in,
                              void* d_out, int out_size, void* d_ws, size_t ws_size,
                              hipStream_t stream) {
    (void)n_in; (void)out_size; (void)ws_size;
    const float* f = (const float*)d_in[0];
    const float* p0 = (const float*)d_in[1];
    const float* v0 = (const float*)d_in[2];
    float* out = (float*)d_out;
    const long K = (long)in_sizes[0] / 3; /* 8388608, multiple of 4096 */
    const int NB = (int)((K + ROWS_PER_BLOCK - 1) / ROWS_PER_BLOCK);

    float* agg = (float*)d_ws; /* NB*6 f32 aggregates */
    size_t off = ((size_t)NB * 6 * sizeof(float) + 255u) & ~(size_t)255u;
    double* carry = (double*)((char*)d_ws + off); /* NB*6 f64 carries */

    phase1_aggregate<<<NB, THREADS, 0, stream>>>(f, agg, K);
    phase2_scan<<<1, THREADS, 0, stream>>>(agg, carry, NB);
    phase3_emit<<<NB, THREADS, 0, stream>>>(f, p0, v0, carry, out, K);
}